// SoftContrastiveLoss_41231686042212
// MI455X (gfx1250) — compile-verified
//
#include <hip/hip_runtime.h>
#include <math.h>

#define ALPHA_C 10.0f
#define BETA_C  2.0f
#define BASE_C  0.7f
#define SELF_TH (1.0f - 1e-6f)
#define BIGV    1e9f

typedef __attribute__((ext_vector_type(2))) float v2f;
typedef __attribute__((ext_vector_type(8))) float v8f;

// ---------- wave reductions (wave32) ----------
__device__ __forceinline__ float wsum16(float v) {
    v += __shfl_xor(v, 1); v += __shfl_xor(v, 2);
    v += __shfl_xor(v, 4); v += __shfl_xor(v, 8);
    return v;
}
__device__ __forceinline__ float wsum32(float v) {
    v = wsum16(v); v += __shfl_xor(v, 16); return v;
}
__device__ __forceinline__ float wmin16(float v) {
    v = fminf(v, __shfl_xor(v, 1)); v = fminf(v, __shfl_xor(v, 2));
    v = fminf(v, __shfl_xor(v, 4)); v = fminf(v, __shfl_xor(v, 8));
    return v;
}
__device__ __forceinline__ float wmax16(float v) {
    v = fmaxf(v, __shfl_xor(v, 1)); v = fmaxf(v, __shfl_xor(v, 2));
    v = fmaxf(v, __shfl_xor(v, 4)); v = fmaxf(v, __shfl_xor(v, 8));
    return v;
}

__device__ __forceinline__ float sigmoidf_(float x) { return 1.0f / (1.0f + expf(-x)); }
// logaddexp(0, z) = log(1+exp(z)), stable
__device__ __forceinline__ float softplus_(float z) {
    return fmaxf(z, 0.0f) + log1pf(expf(-fabsf(z)));
}

// ---------- 16x64 f32 WMMA micro-kernel: 4 adjacent 16x16 tiles ----------
// A: LDS strip (16 rows x d, row stride d). B: columns [col_base, col_base+64) of X.
// Lane m = lane&15, g = lane>>4. A frag: {A[m][k0+2g], A[m][k0+2g+1]}; B frag same
// pattern on the column row-block (B = X^T tile); acc[t][r] = sim[r+8g][16t + m].
__device__ __forceinline__ void mm_tile4(const float* __restrict__ x,
                                         const float* __restrict__ sA,
                                         int d, long col_base, int m, int g,
                                         v8f acc[4]) {
    const float* a  = sA + (long)m * d + 2 * g;
    const float* b0 = x + (col_base + m) * (long)d + 2 * g;
    const float* b1 = b0 + 16L * d;
    const float* b2 = b0 + 32L * d;
    const float* b3 = b0 + 48L * d;
#pragma unroll 2
    for (int k = 0; k < d; k += 4) {
        v2f av  = *(const v2f*)(a + k);
        v2f bv0 = *(const v2f*)(b0 + k);
        v2f bv1 = *(const v2f*)(b1 + k);
        v2f bv2 = *(const v2f*)(b2 + k);
        v2f bv3 = *(const v2f*)(b3 + k);
        acc[0] = __builtin_amdgcn_wmma_f32_16x16x4_f32(false, av, false, bv0, (short)0, acc[0], false, false);
        acc[1] = __builtin_amdgcn_wmma_f32_16x16x4_f32(false, av, false, bv1, (short)0, acc[1], false, false);
        acc[2] = __builtin_amdgcn_wmma_f32_16x16x4_f32(false, av, false, bv2, (short)0, acc[2], false, false);
        acc[3] = __builtin_amdgcn_wmma_f32_16x16x4_f32(false, av, false, bv3, (short)0, acc[3], false, false);
    }
}

// ---------- kernel 0: zero the scalar accumulators ----------
__global__ void kzero(float* sc) {
    if (threadIdx.x < 16) sc[threadIdx.x] = 0.0f;
}

// ---------- kernel A: per-row min_pos / max_neg + length_AP ----------
__global__ void __launch_bounds__(256) kminmax(const float* __restrict__ x,
                                               const int* __restrict__ tg,
                                               float* __restrict__ minpos,
                                               float* __restrict__ maxneg,
                                               float* __restrict__ sc,
                                               int n, int d) {
    __shared__ float sA[16 * 512];
    __shared__ float sMin[8][16];
    __shared__ float sMax[8][16];
    const int tid = threadIdx.x, lane = tid & 31, wave = tid >> 5;
    const int row_base = blockIdx.x * 16;
    const int m = lane & 15, g = lane >> 4;

    const float* src = x + (long)row_base * d;
    for (int i = tid; i < 16 * d; i += 256) sA[i] = src[i];
    __syncthreads();

    int trow[8];
#pragma unroll
    for (int r = 0; r < 8; ++r) trow[r] = tg[row_base + r + 8 * g];

    float minp[8], maxn[8];
#pragma unroll
    for (int r = 0; r < 8; ++r) { minp[r] = BIGV; maxn[r] = -BIGV; }
    float cAP = 0.0f;

    const int ngrp = n >> 6;  // 64-column groups
    for (int cg = wave; cg < ngrp; cg += 8) {
        long col_base = (long)cg * 64;
        v8f acc[4] = {};
        mm_tile4(x, sA, d, col_base, m, g, acc);
#pragma unroll
        for (int t = 0; t < 4; ++t) {
            int tc = tg[col_base + t * 16 + m];
#pragma unroll
            for (int r = 0; r < 8; ++r) {
                float s = acc[t][r];
                bool same = (trow[r] == tc);
                bool pos  = same && (s < SELF_TH);
                if (pos)  { minp[r] = fminf(minp[r], s); cAP += 1.0f; }
                if (!same) { maxn[r] = fmaxf(maxn[r], s); }
            }
        }
    }

#pragma unroll
    for (int r = 0; r < 8; ++r) { minp[r] = wmin16(minp[r]); maxn[r] = wmax16(maxn[r]); }
    if (m == 0) {
#pragma unroll
        for (int r = 0; r < 8; ++r) {
            sMin[wave][r + 8 * g] = minp[r];
            sMax[wave][r + 8 * g] = maxn[r];
        }
    }
    cAP = wsum32(cAP);
    if (lane == 0) atomicAdd(&sc[0], cAP);   // exact integer-valued sum
    __syncthreads();
    if (tid < 16) {
        float mn = BIGV, mx = -BIGV;
#pragma unroll
        for (int w = 0; w < 8; ++w) {
            mn = fminf(mn, sMin[w][tid]);
            mx = fmaxf(mx, sMax[w][tid]);
        }
        minpos[row_base + tid] = mn;   // block owns these rows exclusively
        maxneg[row_base + tid] = mx;
    }
}

// ---------- kernel B: length_an_t with raw margin ----------
__global__ void __launch_bounds__(256) kcount(const float* __restrict__ x,
                                              const int* __restrict__ tg,
                                              const float* __restrict__ minpos,
                                              const float* __restrict__ marginp,
                                              float* __restrict__ sc,
                                              int n, int d) {
    __shared__ float sA[16 * 512];
    const int tid = threadIdx.x, lane = tid & 31, wave = tid >> 5;
    const int row_base = blockIdx.x * 16;
    const int m = lane & 15, g = lane >> 4;

    const float* src = x + (long)row_base * d;
    for (int i = tid; i < 16 * d; i += 256) sA[i] = src[i];
    __syncthreads();

    const float margin = *marginp;
    int trow[8];
    float mp[8];
#pragma unroll
    for (int r = 0; r < 8; ++r) {
        int row = row_base + r + 8 * g;
        trow[r] = tg[row];
        float v = minpos[row];
        mp[r] = (v > 0.5f * BIGV) ? 1.0f : v;  // has_pos fallback
    }

    float cAN = 0.0f;
    const int ngrp = n >> 6;
    for (int cg = wave; cg < ngrp; cg += 8) {
        long col_base = (long)cg * 64;
        v8f acc[4] = {};
        mm_tile4(x, sA, d, col_base, m, g, acc);
#pragma unroll
        for (int t = 0; t < 4; ++t) {
            int tc = tg[col_base + t * 16 + m];
#pragma unroll
            for (int r = 0; r < 8; ++r) {
                float s = acc[t][r];
                if ((trow[r] != tc) && (s + margin - mp[r] > 0.0f)) cAN += 1.0f;
            }
        }
    }
    cAN = wsum32(cAN);
    if (lane == 0) atomicAdd(&sc[1], cAN);   // exact integer-valued sum
}

// ---------- kernel S: derive adjusted margins ----------
__global__ void kscal(const float* __restrict__ marginp, float* __restrict__ sc) {
    if (threadIdx.x == 0) {
        float AP = sc[0], an = sc[1];
        float margin = *marginp;
        float r = an / AP;
        float tol = sigmoidf_(r) * 0.5f;
        float map, man;
        if (r > 1.0f) { map = margin + tol * margin; man = margin * 0.1f - tol * (margin * 0.1f); }
        else          { map = margin;                man = margin * 0.1f; }
        sc[3] = map;
        sc[4] = man;
    }
}

// ---------- kernel C: pass-2 per-row counts and loss sums ----------
__global__ void __launch_bounds__(256) kloss(const float* __restrict__ x,
                                             const int* __restrict__ tg,
                                             const float* __restrict__ minpos,
                                             const float* __restrict__ maxneg,
                                             const float* __restrict__ sc,
                                             float* __restrict__ cntpos,
                                             float* __restrict__ cntneg,
                                             float* __restrict__ sumpos,
                                             float* __restrict__ sumneg,
                                             int n, int d) {
    __shared__ float sA[16 * 512];
    __shared__ float sCp[8][16], sCn[8][16], sSp[8][16], sSn[8][16];
    const int tid = threadIdx.x, lane = tid & 31, wave = tid >> 5;
    const int row_base = blockIdx.x * 16;
    const int m = lane & 15, g = lane >> 4;

    const float* src = x + (long)row_base * d;
    for (int i = tid; i < 16 * d; i += 256) sA[i] = src[i];
    __syncthreads();

    const float map = sc[3];
    const float man = sc[4];
    int trow[8];
    float mp[8], mx[8];
#pragma unroll
    for (int r = 0; r < 8; ++r) {
        int row = row_base + r + 8 * g;
        trow[r] = tg[row];
        float vp = minpos[row];
        float vn = maxneg[row];
        mp[r] = (vp > 0.5f * BIGV) ? 1.0f : vp;
        mx[r] = (vn < -0.5f * BIGV) ? -1.0f : vn;
    }

    float cp[8], cn[8], sp[8], sn[8];
#pragma unroll
    for (int r = 0; r < 8; ++r) { cp[r] = 0; cn[r] = 0; sp[r] = 0; sn[r] = 0; }

    const int ngrp = n >> 6;
    for (int cg = wave; cg < ngrp; cg += 8) {
        long col_base = (long)cg * 64;
        v8f acc[4] = {};
        mm_tile4(x, sA, d, col_base, m, g, acc);
#pragma unroll
        for (int t = 0; t < 4; ++t) {
            int tc = tg[col_base + t * 16 + m];
#pragma unroll
            for (int r = 0; r < 8; ++r) {
                float s = acc[t][r];
                bool same = (trow[r] == tc);
                if (!same && (s + man - mp[r] > 0.0f)) {
                    cn[r] += 1.0f;
                    sn[r] += softplus_(ALPHA_C * (s - BASE_C));
                }
                if (same && (s < SELF_TH) && (mx[r] - s + map > 0.0f)) {
                    cp[r] += 1.0f;
                    sp[r] += softplus_(-BETA_C * (s - BASE_C));
                }
            }
        }
    }

#pragma unroll
    for (int r = 0; r < 8; ++r) {
        cp[r] = wsum16(cp[r]); cn[r] = wsum16(cn[r]);
        sp[r] = wsum16(sp[r]); sn[r] = wsum16(sn[r]);
    }
    if (m == 0) {
#pragma unroll
        for (int r = 0; r < 8; ++r) {
            sCp[wave][r + 8 * g] = cp[r];
            sCn[wave][r + 8 * g] = cn[r];
            sSp[wave][r + 8 * g] = sp[r];
            sSn[wave][r + 8 * g] = sn[r];
        }
    }
    __syncthreads();
    if (tid < 16) {
        float a = 0, b = 0, c = 0, e = 0;
#pragma unroll
        for (int w = 0; w < 8; ++w) {   // fixed order -> deterministic float sums
            a += sCp[w][tid]; b += sCn[w][tid];
            c += sSp[w][tid]; e += sSn[w][tid];
        }
        cntpos[row_base + tid] = a;
        cntneg[row_base + tid] = b;
        sumpos[row_base + tid] = c;
        sumneg[row_base + tid] = e;
    }
}

// ---------- kernel D: final reduction + last-row stats + outputs ----------
__global__ void __launch_bounds__(256) kfinal(const float* __restrict__ x,
                                              const int* __restrict__ tg,
                                              const float* __restrict__ cntpos,
                                              const float* __restrict__ cntneg,
                                              const float* __restrict__ sumpos,
                                              const float* __restrict__ sumneg,
                                              const float* __restrict__ sc,
                                              float* __restrict__ out,
                                              int n, int d) {
    __shared__ float rbuf[8][256];
    __shared__ float xlast[512];
    const int tid = threadIdx.x;

    for (int i = tid; i < d; i += 256) xlast[i] = x[(long)(n - 1) * d + i];
    __syncthreads();

    float lsum = 0, vcnt = 0, lap = 0, lan = 0;
    for (int i = tid; i < n; i += 256) {
        float cp = cntpos[i], cn = cntneg[i];
        bool valid = (cp >= 1.0f) && (cn >= 1.0f);
        float pl = (2.0f / BETA_C) * sumpos[i] / fmaxf(cp, 1.0f);
        float nl = (2.0f / ALPHA_C) * sumneg[i] / fmaxf(cn, 1.0f);
        if (valid) { lsum += pl + nl; vcnt += 1.0f; lap += cp; lan += cn; }
    }

    // last-row mean sims: recompute row n-1 of sim (matvec, trivial cost)
    const int tlast = tg[n - 1];
    float spL = 0, cpL = 0, snL = 0, cnL = 0;
    for (int j = tid; j < n; j += 256) {
        const float* xr = x + (long)j * d;
        float dot = 0.0f;
#pragma unroll 4
        for (int k = 0; k < d; ++k) dot = fmaf(xlast[k], xr[k], dot);
        bool same = (tg[j] == tlast);
        if (same && (dot < SELF_TH)) { spL += dot; cpL += 1.0f; }
        if (!same)                   { snL += dot; cnL += 1.0f; }
    }

    rbuf[0][tid] = lsum; rbuf[1][tid] = vcnt; rbuf[2][tid] = lap; rbuf[3][tid] = lan;
    rbuf[4][tid] = spL;  rbuf[5][tid] = cpL;  rbuf[6][tid] = snL; rbuf[7][tid] = cnL;
    __syncthreads();
    for (int s = 128; s > 0; s >>= 1) {
        if (tid < s) {
#pragma unroll
            for (int q = 0; q < 8; ++q) rbuf[q][tid] += rbuf[q][tid + s];
        }
        __syncthreads();
    }

    if (tid == 0) {
        float fn   = (float)n;
        float loss = rbuf[0][0] / fn;
        float prec = (fn - rbuf[1][0]) / fn;
        float mps  = rbuf[4][0] / fmaxf(rbuf[5][0], 1.0f);
        float mns  = rbuf[6][0] / fmaxf(rbuf[7][0], 1.0f);
        float lp   = rbuf[2][0] * 0.5f;
        float ln   = rbuf[3][0] * 0.5f;
        float Lp   = sc[0] * 0.5f;
        out[0] = loss; out[1] = prec; out[2] = mps; out[3] = mns; out[4] = lp; out[5] = ln;
        float s0 = ln / lp, s2 = ln / Lp, s4 = lp / Lp;
        out[6]  = s0; out[7]  = sigmoidf_(s0); out[8]  = s2; out[9]  = sigmoidf_(s2); out[10] = s4;
        out[11] = s0; out[12] = sigmoidf_(s0); out[13] = s2; out[14] = sigmoidf_(s2); out[15] = s4;
    }
}

extern "C" void kernel_launch(void* const* d_in, const int* in_sizes, int n_in,
                              void* d_out, int out_size, void* d_ws, size_t ws_size,
                              hipStream_t stream) {
    const float* x       = (const float*)d_in[0];
    const int*   tg      = (const int*)d_in[1];
    const float* marginp = (const float*)d_in[2];
    const int n = in_sizes[1];
    const int d = in_sizes[0] / n;

    float* ws     = (float*)d_ws;
    float* minpos = ws;
    float* maxneg = ws + (size_t)n;
    float* cntpos = ws + 2 * (size_t)n;
    float* cntneg = ws + 3 * (size_t)n;
    float* sumpos = ws + 4 * (size_t)n;
    float* sumneg = ws + 5 * (size_t)n;
    float* sc     = ws + 6 * (size_t)n;   // 16 scalar slots
    float* out    = (float*)d_out;

    const int nblk = n / 16;
    kzero<<<1, 32, 0, stream>>>(sc);
    kminmax<<<nblk, 256, 0, stream>>>(x, tg, minpos, maxneg, sc, n, d);
    kcount<<<nblk, 256, 0, stream>>>(x, tg, minpos, marginp, sc, n, d);
    kscal<<<1, 32, 0, stream>>>(marginp, sc);
    kloss<<<nblk, 256, 0, stream>>>(x, tg, minpos, maxneg, sc,
                                    cntpos, cntneg, sumpos, sumneg, n, d);
    kfinal<<<1, 256, 0, stream>>>(x, tg, cntpos, cntneg, sumpos, sumneg, sc, out, n, d);
}